// AttnDecoder_40235253629035
// MI455X (gfx1250) — compile-verified
//
#include <hip/hip_runtime.h>
#include <hip/hip_bf16.h>

// Problem constants
#define VSZ 50257
#define HSZ 2048
#define LSZ 512

typedef __attribute__((ext_vector_type(2))) float v2f;
typedef __attribute__((ext_vector_type(8))) float v8f;

// ---------------------------------------------------------------------------
// Generic 16-rows-per-wave f32 GEMV using V_WMMA_F32_16X16X4_F32.
//   y[r] = act( dot(W[r,:], x) + bias[r] ),  r in [0, rows)
// A = 16x4 weight tile (exact f32), B = x broadcast across N columns.
// Block = 256 threads = 8 waves; each wave owns one 16-row tile.
// Weights are streamed once -> non-temporal loads (don't thrash 192MB L2;
// fc_W alone is 412MB).
// ---------------------------------------------------------------------------
__global__ __launch_bounds__(256) void gemv16_wmma(
    const float* __restrict__ W, const float* __restrict__ x,
    const float* __restrict__ bias, float* __restrict__ y,
    int rows, int K, int relu)
{
    __shared__ float s_x[4096];           // K <= 4096 for all our GEMVs
    const int tid = threadIdx.x;
    for (int i = tid; i < K; i += 256) s_x[i] = x[i];
    __syncthreads();

    const int wave  = tid >> 5;
    const int lane  = tid & 31;
    const int tiles = (rows + 15) >> 4;
    const int tile  = blockIdx.x * 8 + wave;
    if (tile >= tiles) return;            // wave-uniform: EXEC stays all-1s

    const int r0 = tile << 4;
    const int m  = lane & 15;             // A-matrix row (M)
    const int h  = lane >> 4;             // lane half -> K pair select
    int row = r0 + m;
    if (row > rows - 1) row = rows - 1;   // clamp: keep addresses valid
    const v2f* __restrict__ Wp =
        (const v2f*)(W + (size_t)row * (size_t)K + (size_t)(2 * h));

    v8f c = {};
    for (int k0 = 0; k0 < K; k0 += 4) {
        v2f a = __builtin_nontemporal_load(Wp + (k0 >> 1)); // W[row][k0+2h..+1], NT stream
        v2f b;
        b.x = s_x[k0 + 2 * h];            // B[K=2h][n]   broadcast over N
        b.y = s_x[k0 + 2 * h + 1];        // B[K=2h+1][n]
        c = __builtin_amdgcn_wmma_f32_16x16x4_f32(
                false, a, false, b, (short)0, c, false, false);
    }

    // D layout: VGPR v, lane-half h  ->  y[r0 + v + 8h]; N = lane&15 (all equal)
    if (m == 0) {
#pragma unroll
        for (int v = 0; v < 8; ++v) {
            int r = r0 + 8 * h + v;
            if (r < rows) {
                float val = c[v] + bias[r];
                if (relu) val = fmaxf(val, 0.0f);
                y[r] = val;
            }
        }
    }
}

// attn_in = concat(emb_table[x], hidden)                              [2H]
__global__ void setup_attn_in(const int* __restrict__ x,
                              const float* __restrict__ emb,
                              const float* __restrict__ hid,
                              float* __restrict__ attn_in)
{
    int i = blockIdx.x * blockDim.x + threadIdx.x;
    if (i >= 2 * HSZ) return;
    attn_in[i] = (i < HSZ) ? emb[(size_t)x[0] * HSZ + i] : hid[i - HSZ];
}

// softmax over L=512 logits (single block, in-place) + copy to d_out
__global__ __launch_bounds__(512) void softmax512(float* __restrict__ logits,
                                                  float* __restrict__ out_w)
{
    __shared__ float s[512];
    const int t = threadIdx.x;
    float v = logits[t];
    s[t] = v; __syncthreads();
    for (int off = 256; off > 0; off >>= 1) {
        if (t < off) s[t] = fmaxf(s[t], s[t + off]);
        __syncthreads();
    }
    float mx = s[0]; __syncthreads();
    float e = __expf(v - mx);
    s[t] = e; __syncthreads();
    for (int off = 256; off > 0; off >>= 1) {
        if (t < off) s[t] = s[t] + s[t + off];
        __syncthreads();
    }
    float w = e / s[0];
    logits[t] = w;      // ws copy for attn_apply
    out_w[t]  = w;      // final output (attn_weight)
}

// comb_in = concat(emb, attn_weight @ encoder_outputs)                [2H]
__global__ void attn_apply(const float* __restrict__ w,
                           const float* __restrict__ enc,
                           const int* __restrict__ x,
                           const float* __restrict__ emb,
                           float* __restrict__ comb_in)
{
    int i = blockIdx.x * blockDim.x + threadIdx.x;
    if (i >= HSZ) return;
    comb_in[i] = emb[(size_t)x[0] * HSZ + i];
    float acc = 0.0f;
    for (int l = 0; l < LSZ; ++l)
        acc = fmaf(w[l], enc[(size_t)l * HSZ + i], acc);  // coalesced over i
    comb_in[HSZ + i] = acc;
}

// single-step GRU pointwise (PyTorch r,z,n layout)
__global__ void gru_step(const float* __restrict__ gi, const float* __restrict__ gh,
                         const float* __restrict__ h0,
                         float* __restrict__ h1_ws, float* __restrict__ h1_out)
{
    int i = blockIdx.x * blockDim.x + threadIdx.x;
    if (i >= HSZ) return;
    float r = 1.0f / (1.0f + __expf(-(gi[i] + gh[i])));
    float z = 1.0f / (1.0f + __expf(-(gi[HSZ + i] + gh[HSZ + i])));
    float n = tanhf(gi[2 * HSZ + i] + r * gh[2 * HSZ + i]);
    float h = (1.0f - z) * n + z * h0[i];
    h1_ws[i]  = h;
    h1_out[i] = h;
}

// two-pass log-softmax reduction over V (single 1024-thread block; 400KB of
// L2-resident logits -> negligible vs the 412MB fc stream)
__global__ __launch_bounds__(1024) void logsm_reduce(const float* __restrict__ logits,
                                                     float* __restrict__ red, int n)
{
    __shared__ float s[1024];
    const int t = threadIdx.x;
    float m = -3.402823466e38f;
    for (int i = t; i < n; i += 1024) m = fmaxf(m, logits[i]);
    s[t] = m; __syncthreads();
    for (int off = 512; off > 0; off >>= 1) {
        if (t < off) s[t] = fmaxf(s[t], s[t + off]);
        __syncthreads();
    }
    float mx = s[0]; __syncthreads();
    float sum = 0.0f;
    for (int i = t; i < n; i += 1024) sum += __expf(logits[i] - mx);
    s[t] = sum; __syncthreads();
    for (int off = 512; off > 0; off >>= 1) {
        if (t < off) s[t] += s[t + off];
        __syncthreads();
    }
    if (t == 0) { red[0] = mx; red[1] = __logf(s[0]); }
}

__global__ void logsm_write(const float* __restrict__ logits,
                            const float* __restrict__ red,
                            float* __restrict__ out, int n)
{
    int i = blockIdx.x * blockDim.x + threadIdx.x;
    if (i >= n) return;
    out[i] = logits[i] - red[0] - red[1];
}

// ---------------------------------------------------------------------------
// Workspace layout (float offsets)
// ---------------------------------------------------------------------------
#define WS_ATTN_IN 0                      // 2H = 4096
#define WS_ALOG    4096                   // L  = 512  (logits -> weights)
#define WS_COMB_IN 4608                   // 2H = 4096
#define WS_GIN     8704                   // H  = 2048
#define WS_GI      10752                  // 3H = 6144
#define WS_GH      16896                  // 3H = 6144
#define WS_H1      23040                  // H  = 2048
#define WS_LOGITS  25088                  // V  = 50257
#define WS_RED     75345                  // 2

extern "C" void kernel_launch(void* const* d_in, const int* in_sizes, int n_in,
                              void* d_out, int out_size, void* d_ws, size_t ws_size,
                              hipStream_t stream)
{
    const int*   x      = (const int*)  d_in[0];
    const float* hidden = (const float*)d_in[1];   // [H]
    const float* enc    = (const float*)d_in[2];   // [L,H]
    const float* emb    = (const float*)d_in[3];   // [V,H]
    const float* attn_W = (const float*)d_in[4];   // [L,2H]
    const float* attn_b = (const float*)d_in[5];   // [L]
    const float* comb_W = (const float*)d_in[6];   // [H,2H]
    const float* comb_b = (const float*)d_in[7];   // [H]
    const float* W_ih   = (const float*)d_in[8];   // [3H,H]
    const float* W_hh   = (const float*)d_in[9];   // [3H,H]
    const float* b_ih   = (const float*)d_in[10];  // [3H]
    const float* b_hh   = (const float*)d_in[11];  // [3H]
    const float* fc_W   = (const float*)d_in[12];  // [V,H]
    const float* fc_b   = (const float*)d_in[13];  // [V]

    float* out   = (float*)d_out;                  // [V] out, [H] h1, [L] attn_w
    float* ws    = (float*)d_ws;

    float* attn_in = ws + WS_ATTN_IN;
    float* alog    = ws + WS_ALOG;
    float* comb_in = ws + WS_COMB_IN;
    float* g_in    = ws + WS_GIN;
    float* gi      = ws + WS_GI;
    float* gh      = ws + WS_GH;
    float* h1      = ws + WS_H1;
    float* logits  = ws + WS_LOGITS;
    float* red     = ws + WS_RED;

    auto blocks_for = [](int rows) { return (((rows + 15) >> 4) + 7) >> 3; };

    // 1) attn_in = concat(emb[x], h0)
    setup_attn_in<<<(2 * HSZ + 255) / 256, 256, 0, stream>>>(x, emb, hidden, attn_in);
    // 2) attn logits = attn_in @ attn_W.T + attn_b          [L x 2H]
    gemv16_wmma<<<blocks_for(LSZ), 256, 0, stream>>>(attn_W, attn_in, attn_b, alog,
                                                     LSZ, 2 * HSZ, 0);
    // 3) softmax -> attn_weight (ws + d_out tail)
    softmax512<<<1, 512, 0, stream>>>(alog, out + VSZ + HSZ);
    // 4) comb_in = concat(emb[x], attn_weight @ enc)
    attn_apply<<<(HSZ + 255) / 256, 256, 0, stream>>>(alog, enc, x, emb, comb_in);
    // 5) g_in = relu(comb_in @ comb_W.T + comb_b)           [H x 2H]
    gemv16_wmma<<<blocks_for(HSZ), 256, 0, stream>>>(comb_W, comb_in, comb_b, g_in,
                                                     HSZ, 2 * HSZ, 1);
    // 6) gi = g_in @ W_ih.T + b_ih ; gh = h0 @ W_hh.T + b_hh  [3H x H]
    gemv16_wmma<<<blocks_for(3 * HSZ), 256, 0, stream>>>(W_ih, g_in, b_ih, gi,
                                                         3 * HSZ, HSZ, 0);
    gemv16_wmma<<<blocks_for(3 * HSZ), 256, 0, stream>>>(W_hh, hidden, b_hh, gh,
                                                         3 * HSZ, HSZ, 0);
    // 7) GRU pointwise -> h1 (ws + d_out middle)
    gru_step<<<(HSZ + 255) / 256, 256, 0, stream>>>(gi, gh, hidden, h1, out + VSZ);
    // 8) logits = h1 @ fc_W.T + fc_b                        [V x H]  (dominant)
    gemv16_wmma<<<blocks_for(VSZ), 256, 0, stream>>>(fc_W, h1, fc_b, logits,
                                                     VSZ, HSZ, 0);
    // 9) log-softmax
    logsm_reduce<<<1, 1024, 0, stream>>>(logits, red, VSZ);
    logsm_write<<<(VSZ + 255) / 256, 256, 0, stream>>>(logits, red, out, VSZ);

    (void)in_sizes; (void)n_in; (void)out_size; (void)ws_size;
}